// CompressiveKV_52544629899299
// MI455X (gfx1250) — compile-verified
//
#include <hip/hip_runtime.h>
#include <hip/hip_bf16.h>
#include <math.h>

typedef float v2f __attribute__((ext_vector_type(2)));
typedef float v4f __attribute__((ext_vector_type(4)));
typedef float v8f __attribute__((ext_vector_type(8)));

#define LN_EPS 1e-5f
#define NROWS   1024      // B*H*SLOTS = 4*32*8
#define DL      128
#define SEGLEN  1024      // T / SLOTS

__device__ __forceinline__ float wave_sum(float x) {
#pragma unroll
  for (int off = 16; off > 0; off >>= 1) x += __shfl_xor(x, off, 32);
  return x;
}

// ---------- Kernel 1: segment mean pool (streams 1 GiB; HBM-bound) ----------
__global__ void ckv_pool(const float* __restrict__ k_lat,
                         const float* __restrict__ v_lat,
                         float* __restrict__ kseg,
                         float* __restrict__ vseg) {
  const int task = blockIdx.x;
  const float* __restrict__ src = (task & 1) ? v_lat : k_lat;
  float* __restrict__       dst = (task & 1) ? vseg  : kseg;
  const int seg = task >> 1;

  const v4f* __restrict__ base =
      (const v4f*)src + (size_t)seg * (size_t)SEGLEN * (DL / 4);
  const int lane4 = threadIdx.x & 31;
  const int rowg  = threadIdx.x >> 5;

  v4f acc = {0.f, 0.f, 0.f, 0.f};
  for (int row = rowg; row < SEGLEN; row += 8) {
    v4f x = __builtin_nontemporal_load(base + row * (DL / 4) + lane4);
    acc += x;
  }

  __shared__ v4f red[8][32];
  red[rowg][lane4] = acc;
  __syncthreads();

  if (rowg == 0) {
    v4f s = red[0][lane4];
#pragma unroll
    for (int g = 1; g < 8; ++g) s += red[g][lane4];
    s *= (1.0f / (float)SEGLEN);
    ((v4f*)dst)[seg * (DL / 4) + lane4] = s;
  }
}

// ---------- Kernel 2: gates + LNs, one wave32 per row ----------
__device__ __forceinline__ void ln128(const float (&x)[4],
                                      const float* __restrict__ g,
                                      const float* __restrict__ b,
                                      int lane, float (&y)[4]) {
  float s = x[0] + x[1] + x[2] + x[3];
  float q = x[0]*x[0] + x[1]*x[1] + x[2]*x[2] + x[3]*x[3];
  float mean = wave_sum(s) * (1.0f / (float)DL);
  float msq  = wave_sum(q) * (1.0f / (float)DL);
  float rstd = rsqrtf(msq - mean * mean + LN_EPS);
#pragma unroll
  for (int i = 0; i < 4; ++i)
    y[i] = (x[i] - mean) * rstd * g[lane * 4 + i] + b[lane * 4 + i];
}

__global__ void ckv_head(const float* __restrict__ kseg,
                         const float* __restrict__ vseg,
                         const float* __restrict__ wg_ln_g, const float* __restrict__ wg_ln_b,
                         const float* __restrict__ wg_w,    const float* __restrict__ wg_b,
                         const float* __restrict__ rh_ln_g, const float* __restrict__ rh_ln_b,
                         const float* __restrict__ rh_w,    const float* __restrict__ rh_b,
                         const float* __restrict__ pk_ln_g, const float* __restrict__ pk_ln_b,
                         const float* __restrict__ pv_ln_g, const float* __restrict__ pv_ln_b,
                         float* __restrict__ kn, float* __restrict__ vn) {
  const int lane = threadIdx.x & 31;
  const int row  = blockIdx.x * 8 + (threadIdx.x >> 5);

  v4f kv4 = ((const v4f*)kseg)[row * (DL / 4) + lane];
  v4f vv4 = ((const v4f*)vseg)[row * (DL / 4) + lane];
  float k[4], v[4], xh[4];
#pragma unroll
  for (int i = 0; i < 4; ++i) { k[i] = kv4[i]; v[i] = vv4[i]; }

  // write gate
  ln128(k, wg_ln_g, wg_ln_b, lane, xh);
  float d = 0.f;
#pragma unroll
  for (int i = 0; i < 4; ++i) d += xh[i] * wg_w[lane * 4 + i];
  float logit = wave_sum(d) + wg_b[0];
  float g = 1.0f / (1.0f + __expf(-logit));
#pragma unroll
  for (int i = 0; i < 4; ++i) { k[i] *= g; v[i] *= g; }

  // retention head
  ln128(k, rh_ln_g, rh_ln_b, lane, xh);
  float l[3];
#pragma unroll
  for (int j = 0; j < 3; ++j) {
    float p = 0.f;
#pragma unroll
    for (int i = 0; i < 4; ++i) p += xh[i] * rh_w[j * DL + lane * 4 + i];
    l[j] = wave_sum(p) + rh_b[j];
  }
  float mx = fmaxf(l[0], fmaxf(l[1], l[2]));
  float e0 = __expf(l[0] - mx), e1 = __expf(l[1] - mx), e2 = __expf(l[2] - mx);
  float scale = (e0 + 0.5f * e1) / (e0 + e1 + e2);
#pragma unroll
  for (int i = 0; i < 4; ++i) { k[i] *= scale; v[i] *= scale; }

  // post LNs -> inputs for the WMMA GEMM
  float yk[4], yv[4];
  ln128(k, pk_ln_g, pk_ln_b, lane, yk);
  ln128(v, pv_ln_g, pv_ln_b, lane, yv);
  v4f ok, ov;
#pragma unroll
  for (int i = 0; i < 4; ++i) { ok[i] = yk[i]; ov[i] = yv[i]; }
  ((v4f*)kn)[row * (DL / 4) + lane] = ok;
  ((v4f*)vn)[row * (DL / 4) + lane] = ov;
}

// ---------- Kernel 3: out = X @ W^T via V_WMMA_F32_16X16X4_F32 ----------
__global__ void ckv_gemm(const float* __restrict__ kn,
                         const float* __restrict__ vn,
                         const float* __restrict__ pk_w,
                         const float* __restrict__ pv_w,
                         float* __restrict__ out) {
  const int lane = threadIdx.x & 31;
  const int tile = blockIdx.x * 8 + (threadIdx.x >> 5);  // 1024 tiles
  const int which = tile >> 9;
  const int t  = tile & 511;
  const int rt = t >> 3;                // 0..63
  const int ct = t & 7;                 // 0..7

  const float* __restrict__ X = which ? vn   : kn;
  const float* __restrict__ W = which ? pv_w : pk_w;
  float* __restrict__ O = out + (size_t)which * (size_t)(NROWS * DL);

  const int m  = lane & 15;
  const int hi = lane >> 4;
  const float* __restrict__ arow = X + (size_t)(rt * 16 + m) * DL + 2 * hi;
  const float* __restrict__ brow = W + (size_t)(ct * 16 + m) * DL + 2 * hi;

  v8f c = {0.f, 0.f, 0.f, 0.f, 0.f, 0.f, 0.f, 0.f};
#pragma unroll
  for (int kk = 0; kk < DL; kk += 4) {
    v2f a = *(const v2f*)(arow + kk);
    v2f b = *(const v2f*)(brow + kk);
    c = __builtin_amdgcn_wmma_f32_16x16x4_f32(false, a, false, b,
                                              (short)0, c, false, false);
  }

#pragma unroll
  for (int r = 0; r < 8; ++r)
    O[(size_t)(rt * 16 + r + 8 * hi) * DL + ct * 16 + m] = c[r];
}

extern "C" void kernel_launch(void* const* d_in, const int* in_sizes, int n_in,
                              void* d_out, int out_size, void* d_ws, size_t ws_size,
                              hipStream_t stream) {
  const float* k_lat   = (const float*)d_in[0];
  const float* v_lat   = (const float*)d_in[1];
  const float* wg_ln_g = (const float*)d_in[2];
  const float* wg_ln_b = (const float*)d_in[3];
  const float* wg_w    = (const float*)d_in[4];
  const float* wg_b    = (const float*)d_in[5];
  const float* rh_ln_g = (const float*)d_in[6];
  const float* rh_ln_b = (const float*)d_in[7];
  const float* rh_w    = (const float*)d_in[8];
  const float* rh_b    = (const float*)d_in[9];
  const float* pk_ln_g = (const float*)d_in[10];
  const float* pk_ln_b = (const float*)d_in[11];
  const float* pk_w    = (const float*)d_in[12];
  const float* pv_ln_g = (const float*)d_in[13];
  const float* pv_ln_b = (const float*)d_in[14];
  const float* pv_w    = (const float*)d_in[15];

  float* ws   = (float*)d_ws;
  float* kseg = ws;
  float* vseg = ws + (size_t)NROWS * DL;
  float* knrm = ws + (size_t)2 * NROWS * DL;
  float* vnrm = ws + (size_t)3 * NROWS * DL;
  float* out  = (float*)d_out;

  ckv_pool<<<2 * NROWS, 256, 0, stream>>>(k_lat, v_lat, kseg, vseg);
  ckv_head<<<NROWS / 8, 256, 0, stream>>>(kseg, vseg,
                                          wg_ln_g, wg_ln_b, wg_w, wg_b,
                                          rh_ln_g, rh_ln_b, rh_w, rh_b,
                                          pk_ln_g, pk_ln_b, pv_ln_g, pv_ln_b,
                                          knrm, vnrm);
  ckv_gemm<<<(2 * (NROWS / 16) * (DL / 16)) / 8, 256, 0, stream>>>(
      knrm, vnrm, pk_w, pv_w, out);
}